// FastWeight_15762529976380
// MI455X (gfx1250) — compile-verified
//
#include <hip/hip_runtime.h>

typedef float v2f __attribute__((ext_vector_type(2)));
typedef float v4f __attribute__((ext_vector_type(4)));
typedef float v8f __attribute__((ext_vector_type(8)));

#define LAM 0.9f
#define ETA 0.5f

// One workgroup (128 threads = 4 waves) handles a 16-row x H-col tile of one
// batch's A.  Streams A -> A_ with coalesced b128 loads + non-temporal b128
// stores (bandwidth-bound part), and computes x_ = A_ @ x with
// V_WMMA_F32_16X16X4_F32 on LDS-staged chunks (compute is free at this
// arithmetic intensity).
__global__ __launch_bounds__(128) void fastweight_kernel(
    const float* __restrict__ A, const float* __restrict__ x,
    float* __restrict__ Aout, float* __restrict__ xout, int H)
{
    __shared__ float xs[1024];        // x[b][:]
    __shared__ float tileA[16 * 68];  // 16x64 A_ chunk, row stride 68 (bank-skewed, 16B aligned)
    __shared__ float red[64];         // cross-wave mat-vec partials

    const int tid  = threadIdx.x;
    const int wave = tid >> 5;
    const int lane = tid & 31;
    const int b    = blockIdx.y;
    const int rt   = blockIdx.x;      // 16-row tile index

    // stage x[b] into LDS (coalesced)
    const float* xb = x + (size_t)b * H;
    for (int i = tid; i < H; i += 128) xs[i] = xb[i];
    __syncthreads();

    const size_t Abase = ((size_t)b * H + (size_t)rt * 16) * (size_t)H;
    const int row  = tid >> 3;        // 0..15 local row
    const int csub = (tid & 7) * 8;   // 0..56 column offset within chunk
    const float exi = ETA * xs[rt * 16 + row];

    // WMMA operand geometry (per ISA 16x16x4 f32 layout):
    // lane L holds row M = L%16; VGPR0/1 carry K = {0,1} (lanes 0-15) or {2,3} (lanes 16-31)
    const int m    = lane & 15;
    const int koff = (lane < 16) ? 0 : 2;

    v8f cacc = {};                    // chained C/D accumulator (16x16 f32)

    for (int c0 = 0; c0 < H; c0 += 64) {
        const float* ap = A + Abase + (size_t)row * H + c0 + csub;
        v4f a0 = *(const v4f*)ap;
        v4f a1 = *(const v4f*)(ap + 4);
        if (c0 + 64 < H) __builtin_prefetch(ap + 64, 0, 1);   // global_prefetch_b8

        v4f w0, w1;
        #pragma unroll
        for (int j = 0; j < 4; ++j) {
            w0[j] = LAM * a0[j] + exi * xs[c0 + csub + j];
            w1[j] = LAM * a1[j] + exi * xs[c0 + csub + 4 + j];
        }

        // streaming store of A_ (write-once data: non-temporal)
        float* op = Aout + Abase + (size_t)row * H + c0 + csub;
        __builtin_nontemporal_store(w0, (v4f*)op);
        __builtin_nontemporal_store(w1, (v4f*)(op + 4));

        // stage A_ chunk into LDS for the WMMA mat-vec
        *(v4f*)&tileA[row * 68 + csub]     = w0;
        *(v4f*)&tileA[row * 68 + csub + 4] = w1;
        __syncthreads();

        // each wave covers a disjoint 16-wide K slice of this 64-col chunk
        #pragma unroll
        for (int kk = 0; kk < 4; ++kk) {
            const int kl = wave * 16 + kk * 4;
            v2f aop, bop;
            aop.x = tileA[m * 68 + kl + koff];
            aop.y = tileA[m * 68 + kl + koff + 1];
            // broadcast x segment across all 16 N columns
            bop.x = xs[c0 + kl + koff];
            bop.y = xs[c0 + kl + koff + 1];
            cacc = __builtin_amdgcn_wmma_f32_16x16x4_f32(
                false, aop, false, bop, (short)0, cacc, false, false);
        }
        __syncthreads();
    }

    // D columns are all identical (B was a broadcast); column 0 lives in
    // lane 0 (rows 0-7) and lane 16 (rows 8-15) of each wave.
    #pragma unroll
    for (int j = 0; j < 8; ++j) {
        if (lane == 0)  red[wave * 16 + j]     = cacc[j];
        if (lane == 16) red[wave * 16 + 8 + j] = cacc[j];
    }
    __syncthreads();
    if (tid < 16) {
        float s = red[tid] + red[16 + tid] + red[32 + tid] + red[48 + tid];
        xout[(size_t)b * H + (size_t)rt * 16 + tid] = s;
    }
}

extern "C" void kernel_launch(void* const* d_in, const int* in_sizes, int n_in,
                              void* d_out, int out_size, void* d_ws, size_t ws_size,
                              hipStream_t stream) {
    const float* A = (const float*)d_in[0];
    const float* x = (const float*)d_in[1];
    const long long nA = in_sizes[0];   // B*H*H
    const long long nx = in_sizes[1];   // B*H
    const int H = (int)(nA / nx);
    const int B = (int)(nx / H);

    float* Aout = (float*)d_out;
    float* xout = Aout + (size_t)B * (size_t)H * (size_t)H;

    dim3 grid(H / 16, B);
    fastweight_kernel<<<grid, 128, 0, stream>>>(A, x, Aout, xout, H);
}